// LigerLMHeadJSD_56092272886459
// MI455X (gfx1250) — compile-verified
//
#include <hip/hip_runtime.h>
#include <math.h>

// ---------------- problem constants (match reference) ----------------
#define N_TOK 2048
#define HS 2048              // student hidden (K for student GEMM)
#define HT 4096              // teacher hidden
#define VOCAB 32000
#define VSPLIT 10
#define VCHUNK (VOCAB / VSPLIT)          // 3200
#define ROWS 16                          // token rows per block
#define WAVES_PER_BLOCK 8                // 256 threads, wave32
#define TILE_COLS (WAVES_PER_BLOCK * 16) // 128 vocab cols per block-iter
#define IGNORE_INDEX (-100)

typedef float v2f __attribute__((ext_vector_type(2)));
typedef float v8f __attribute__((ext_vector_type(8)));
typedef unsigned int v4u __attribute__((ext_vector_type(4)));
typedef int v4i __attribute__((ext_vector_type(4)));
typedef int v8i __attribute__((ext_vector_type(8)));

// V_WMMA_F32_16X16X4_F32 : A 16x4 f32 (2 VGPR), B 4x16 f32 (2 VGPR), C/D 16x16 f32 (8 VGPR)
static __device__ __forceinline__ v8f wmma_f32_16x16x4(v2f a, v2f b, v8f c) {
  return __builtin_amdgcn_wmma_f32_16x16x4_f32(false, a, false, b, (short)0, c, false, false);
}

// Two WMMAs fed by one contiguous float4 per lane for A and B.
// K-slot permutation trick: lanes 0-15 hold k..k+3, lanes 16-31 hold k+4..k+7.
// WMMA0 consumes elems {0,1}, WMMA1 elems {2,3} of each lane's float4; A and B
// use the same permutation of K, so the dot-product sum is unchanged.
static __device__ __forceinline__ void wmma2_f4(const float4 av, const float4 bv,
                                                v8f& c0, v8f& c1) {
  v2f a0 = {av.x, av.y}, a1 = {av.z, av.w};
  v2f b0 = {bv.x, bv.y}, b1 = {bv.z, bv.w};
  c0 = wmma_f32_16x16x4(a0, b0, c0);
  c1 = wmma_f32_16x16x4(a1, b1, c1);
}

// online logsumexp merge: (M,S) <- merge (m,s)
static __device__ __forceinline__ void olmerge(float& M, float& S, float m, float s) {
  if (m > M) { S = S * __expf(M - m) + s; M = m; }
  else if (m != -INFINITY) { S += s * __expf(m - M); }
}

// ---------------- Tensor Data Mover: 2D tile (rows x rowElems f32, stride rowElems) -> LDS ----------------
static __device__ __forceinline__ void tdm_load_2d(void* ldsDst, const void* gsrc,
                                                   unsigned rowElems, unsigned rows,
                                                   unsigned tensorRows) {
  unsigned long long ga = (unsigned long long)(uintptr_t)gsrc;
  v4u g0;
  g0.x = 1u;                                       // count=1, user descriptor, no gather
  g0.y = (unsigned)(uintptr_t)ldsDst;              // lds byte address
  g0.z = (unsigned)ga;                             // global_addr[31:0]  (bits 95:64)
  g0.w = (unsigned)((ga >> 32) & 0x01FFFFFFu)      // global_addr[56:32]
         | 0x80000000u;                            // type=2 ("image") at bits 127:126
  v8i g1;
  g1[0] = (int)(2u << 16);                                   // data_size=4B, wg_mask=0, no flags
  g1[1] = (int)((rowElems & 0xFFFFu) << 16);                 // tensor_dim0[15:0] @ bits 63:48
  g1[2] = (int)((rowElems >> 16) | ((tensorRows & 0xFFFFu) << 16)); // dim0 hi16 | tensor_dim1 lo16
  g1[3] = (int)((tensorRows >> 16) | ((rowElems & 0xFFFFu) << 16)); // dim1 hi16 | tile_dim0
  g1[4] = (int)(rows & 0xFFFFu);                             // tile_dim1=rows, tile_dim2=0
  g1[5] = (int)rowElems;                                     // tensor_dim0_stride lo32
  g1[6] = 0;                                                 // stride hi16 | dim1_stride lo16
  g1[7] = 0;
  v4i z4 = {0, 0, 0, 0};
#if defined(__clang_major__) && (__clang_major__ >= 23)
  v8i z8 = {0, 0, 0, 0, 0, 0, 0, 0};
  __builtin_amdgcn_tensor_load_to_lds(g0, g1, z4, z4, z8, 0);
#else
  __builtin_amdgcn_tensor_load_to_lds(g0, g1, z4, z4, 0);
#endif
}

// ---------------- pass 1: GEMM + online softmax stats (T=1 and T=2), optional f16 logit cache ----------------
template <bool STAGE_A, bool STORE_Z>
__global__ void gemm_stats_kernel(const float* __restrict__ inp,
                                  const float* __restrict__ wgt,
                                  const float* __restrict__ bias, int K,
                                  float* __restrict__ oM1, float* __restrict__ oS1,
                                  float* __restrict__ oM2, float* __restrict__ oS2,
                                  _Float16* __restrict__ zout) {
  extern __shared__ float lds[];   // STAGE_A: ROWS*K floats (A tile); always >=32KB merge buffer
  const int row0 = blockIdx.x * ROWS;
  const int vc = blockIdx.y;
  const int tid = threadIdx.x;
  const int wave = tid >> 5;
  const int lane = tid & 31;
  const int nm = lane & 15;            // A: row index; B: col index
  const int k4 = (lane >> 4) << 2;     // upper half-wave takes k+4..k+7 of each 8-group
  const int hi = (lane >> 4) << 3;     // C element j -> row j + hi

  if constexpr (STAGE_A) {
    // TDM DMA: 16 x K f32 tile -> LDS; one wave issues, all wait at the barrier
    if (wave == 0) {
      tdm_load_2d(lds, inp + (size_t)row0 * K, (unsigned)K, ROWS, N_TOK);
      __builtin_amdgcn_s_wait_tensorcnt(0);
    }
    __syncthreads();
  }

  float m1[8], s1[8], m2[8], s2[8];
#pragma unroll
  for (int j = 0; j < 8; ++j) { m1[j] = -INFINITY; s1[j] = 0.f; m2[j] = -INFINITY; s2[j] = 0.f; }

  const float* aRow = STAGE_A ? &lds[nm * K] : inp + (size_t)(row0 + nm) * K;
  for (int it = 0; it < VCHUNK / TILE_COLS; ++it) {
    const int col0 = vc * VCHUNK + it * TILE_COLS + wave * 16;
    const float* bRow = wgt + (size_t)(col0 + nm) * K;
    __builtin_prefetch(bRow + (size_t)TILE_COLS * K, 0, 1);  // next tile's weight row
    v8f acc0 = {}, acc1 = {}, acc2 = {}, acc3 = {};
    for (int k = 0; k < K; k += 16) {
      float4 av0 = *(const float4*)&aRow[k + k4];
      float4 bv0 = *(const float4*)&bRow[k + k4];
      float4 av1 = *(const float4*)&aRow[k + 8 + k4];
      float4 bv1 = *(const float4*)&bRow[k + 8 + k4];
      wmma2_f4(av0, bv0, acc0, acc1);
      wmma2_f4(av1, bv1, acc2, acc3);
    }
    const float bc = bias[col0 + nm];
#pragma unroll
    for (int j = 0; j < 8; ++j) {
      float z = acc0[j] + acc1[j] + acc2[j] + acc3[j] + bc;  // logit (row j+hi, col nm)
      olmerge(m1[j], s1[j], z, 1.0f);          // T = 1 stats
      olmerge(m2[j], s2[j], z * 0.5f, 1.0f);   // T = 2 stats
      if constexpr (STORE_Z)
        zout[(size_t)(row0 + j + hi) * VOCAB + (size_t)(col0 + nm)] = (_Float16)z;
    }
  }

  // merge per-lane partials across the block (LDS reuse; 32KB)
  __syncthreads();
  float* buf = lds;
  {
    const int base = (wave * 32 + lane) * 32;
#pragma unroll
    for (int j = 0; j < 8; ++j) {
      buf[base + j * 4 + 0] = m1[j];
      buf[base + j * 4 + 1] = s1[j];
      buf[base + j * 4 + 2] = m2[j];
      buf[base + j * 4 + 3] = s2[j];
    }
  }
  __syncthreads();
  if (tid < ROWS) {
    const int r = tid, j = r & 7, l0 = (r < 8) ? 0 : 16;  // C layout: VGPR j = row j / j+8
    float M1 = -INFINITY, S1 = 0.f, M2 = -INFINITY, S2 = 0.f;
    for (int wv = 0; wv < WAVES_PER_BLOCK; ++wv)
      for (int l = l0; l < l0 + 16; ++l) {
        const float* p = &buf[(wv * 32 + l) * 32 + j * 4];
        olmerge(M1, S1, p[0], p[1]);
        olmerge(M2, S2, p[2], p[3]);
      }
    const size_t o = (size_t)(row0 + r) * VSPLIT + vc;
    oM1[o] = M1; oS1[o] = S1; oM2[o] = M2; oS2[o] = S2;
  }
}

// ---------------- combine chunk partials -> per-row lse ----------------
__global__ void reduce_lse_kernel(const float* __restrict__ pM, const float* __restrict__ pS,
                                  float* __restrict__ lse) {
  const int r = blockIdx.x * blockDim.x + threadIdx.x;
  if (r >= N_TOK) return;
  float M = -INFINITY, S = 0.f;
  for (int c = 0; c < VSPLIT; ++c)
    olmerge(M, S, pM[(size_t)r * VSPLIT + c], pS[(size_t)r * VSPLIT + c]);
  lse[r] = M + __logf(S);
}

// ---------------- hard CE: nll = lse1 - z_target ----------------
__global__ void ce_kernel(const float* __restrict__ inp, const float* __restrict__ wgt,
                          const float* __restrict__ bias, const int* __restrict__ tgt,
                          const float* __restrict__ lse1, float* __restrict__ hard_acc) {
  __shared__ float red[256];
  const int row = blockIdx.x;
  const int t = tgt[row];
  if (t == IGNORE_INDEX) return;
  float p = 0.f;
  const float* a = inp + (size_t)row * HS;
  const float* w = wgt + (size_t)t * HS;
  for (int k = threadIdx.x; k < HS; k += 256) p = fmaf(a[k], w[k], p);
  red[threadIdx.x] = p;
  __syncthreads();
  for (int s = 128; s > 0; s >>= 1) {
    if (threadIdx.x < s) red[threadIdx.x] += red[threadIdx.x + s];
    __syncthreads();
  }
  if (threadIdx.x == 0) atomicAdd(hard_acc, lse1[row] - (red[0] + bias[t]));
}

// ---------------- pass 2a (small ws): recompute both logit tiles with WMMA, accumulate JSD ----------------
__global__ void jsd_kernel(const float* __restrict__ sin_, const float* __restrict__ swgt,
                           const float* __restrict__ sbias,
                           const float* __restrict__ tin_, const float* __restrict__ twgt,
                           const float* __restrict__ tbias,
                           const float* __restrict__ lse2s, const float* __restrict__ lse2t,
                           float* __restrict__ jsd_acc) {
  extern __shared__ float lds[];     // ROWS*HS floats: student A tile
  __shared__ float lsS[ROWS], lsT[ROWS];
  const int row0 = blockIdx.x * ROWS;
  const int vc = blockIdx.y;
  const int tid = threadIdx.x;
  const int wave = tid >> 5, lane = tid & 31;
  const int nm = lane & 15;
  const int k4 = (lane >> 4) << 2;
  const int hi = (lane >> 4) << 3;   // C element j -> row j + hi

  if (wave == 0) {
    tdm_load_2d(lds, sin_ + (size_t)row0 * HS, HS, ROWS, N_TOK);
    __builtin_amdgcn_s_wait_tensorcnt(0);
  }
  if (tid < ROWS) { lsS[tid] = lse2s[row0 + tid]; lsT[tid] = lse2t[row0 + tid]; }
  __syncthreads();

  const float* aS = &lds[nm * HS];
  const float* aT = tin_ + (size_t)(row0 + nm) * HT;   // teacher acts: L2-resident

  float local = 0.f;
  for (int it = 0; it < VCHUNK / TILE_COLS; ++it) {
    const int col0 = vc * VCHUNK + it * TILE_COLS + wave * 16;
    const float* bS = swgt + (size_t)(col0 + nm) * HS;
    const float* bT = twgt + (size_t)(col0 + nm) * HT;
    __builtin_prefetch(bS + (size_t)TILE_COLS * HS, 0, 1);
    __builtin_prefetch(bT + (size_t)TILE_COLS * HT, 0, 1);
    v8f sa0 = {}, sa1 = {}, sa2 = {}, sa3 = {};
    for (int k = 0; k < HS; k += 16) {
      float4 av0 = *(const float4*)&aS[k + k4];
      float4 bv0 = *(const float4*)&bS[k + k4];
      float4 av1 = *(const float4*)&aS[k + 8 + k4];
      float4 bv1 = *(const float4*)&bS[k + 8 + k4];
      wmma2_f4(av0, bv0, sa0, sa1);
      wmma2_f4(av1, bv1, sa2, sa3);
    }
    v8f ta0 = {}, ta1 = {}, ta2 = {}, ta3 = {};
    for (int k = 0; k < HT; k += 16) {
      float4 av0 = *(const float4*)&aT[k + k4];
      float4 bv0 = *(const float4*)&bT[k + k4];
      float4 av1 = *(const float4*)&aT[k + 8 + k4];
      float4 bv1 = *(const float4*)&bT[k + 8 + k4];
      wmma2_f4(av0, bv0, ta0, ta1);
      wmma2_f4(av1, bv1, ta2, ta3);
    }
    const float bcS = sbias[col0 + nm], bcT = tbias[col0 + nm];
#pragma unroll
    for (int j = 0; j < 8; ++j) {
      const int r = j + hi;
      float sp = (sa0[j] + sa1[j] + sa2[j] + sa3[j] + bcS) * 0.5f - lsS[r];  // log p_s (T=2)
      float tp = (ta0[j] + ta1[j] + ta2[j] + ta3[j] + bcT) * 0.5f - lsT[r];  // log p_t (T=2)
      float pa = __expf(sp), pb = __expf(tp);
      float lm = __logf(0.5f * (pa + pb));
      local += 0.5f * (pa * (sp - lm) + pb * (tp - lm));   // beta=0.5 JSD term
    }
  }
#pragma unroll
  for (int off = 16; off > 0; off >>= 1) local += __shfl_xor(local, off, 32);
  if (lane == 0) atomicAdd(jsd_acc, local);
}

// ---------------- pass 2b (large ws): stream cached f16 logits, accumulate JSD ----------------
__global__ void jsd_cached_kernel(const _Float16* __restrict__ sl, const _Float16* __restrict__ tl,
                                  const float* __restrict__ lse2s, const float* __restrict__ lse2t,
                                  float* __restrict__ jsd_acc) {
  __shared__ float wsum[8];
  const int row = blockIdx.x;
  const float Ls = lse2s[row], Lt = lse2t[row];
  const size_t base = (size_t)row * VOCAB;
  float local = 0.f;
  for (int c = threadIdx.x; c < VOCAB; c += 256) {
    float sp = 0.5f * (float)sl[base + c] - Ls;
    float tp = 0.5f * (float)tl[base + c] - Lt;
    float pa = __expf(sp), pb = __expf(tp);
    float lm = __logf(0.5f * (pa + pb));
    local += 0.5f * (pa * (sp - lm) + pb * (tp - lm));
  }
#pragma unroll
  for (int off = 16; off > 0; off >>= 1) local += __shfl_xor(local, off, 32);
  if ((threadIdx.x & 31) == 0) wsum[threadIdx.x >> 5] = local;
  __syncthreads();
  if (threadIdx.x == 0) {
    float t = 0.f;
    for (int i = 0; i < 8; ++i) t += wsum[i];
    atomicAdd(jsd_acc, t);
  }
}

// ---------------- init / finalize ----------------
__global__ void init_acc_kernel(float* acc) { if (threadIdx.x < 2) acc[threadIdx.x] = 0.f; }

__global__ void finalize_kernel(const float* __restrict__ acc, float* __restrict__ out) {
  out[0] = 0.5f * acc[0] / (float)N_TOK + 0.5f * acc[1] / (float)N_TOK;  // W_HARD=W_SOFT=0.5
}

// ---------------- host launcher ----------------
extern "C" void kernel_launch(void* const* d_in, const int* in_sizes, int n_in,
                              void* d_out, int out_size, void* d_ws, size_t ws_size,
                              hipStream_t stream) {
  const float* s_in = (const float*)d_in[0];
  const float* s_w  = (const float*)d_in[1];
  const float* s_b  = (const float*)d_in[2];
  const float* t_in = (const float*)d_in[3];
  const float* t_w  = (const float*)d_in[4];
  const float* t_b  = (const float*)d_in[5];
  const int*   tgt  = (const int*)d_in[6];

  float* ws = (float*)d_ws;
  const size_t P = (size_t)N_TOK * VSPLIT;
  float* sM1 = ws;       float* sS1 = sM1 + P;
  float* sM2 = sS1 + P;  float* sS2 = sM2 + P;
  float* tM1 = sS2 + P;  float* tS1 = tM1 + P;
  float* tM2 = tS1 + P;  float* tS2 = tM2 + P;
  float* lse1s = tS2 + P;
  float* lse2s = lse1s + N_TOK;
  float* lse2t = lse2s + N_TOK;
  float* acc   = lse2t + N_TOK;  // acc[0]=hard sum, acc[1]=jsd sum

  // optional f16 logit cache (saves the second weight sweep if ws is big enough)
  const size_t statsBytes = (8 * P + 3 * N_TOK + 2) * sizeof(float);
  const size_t logitBytes = 2ull * N_TOK * VOCAB * sizeof(_Float16);  // ~262 MB
  const bool useCache = ws_size >= statsBytes + logitBytes + 256;
  _Float16* sLog = (_Float16*)(acc + 2);
  _Float16* tLog = sLog + (size_t)N_TOK * VOCAB;

  init_acc_kernel<<<1, 32, 0, stream>>>(acc);

  dim3 grid(N_TOK / ROWS, VSPLIT);  // 128 x 10
  const size_t ldsS = (size_t)ROWS * HS * sizeof(float);                 // 128 KB (student, staged)
  const size_t ldsM = (size_t)WAVES_PER_BLOCK * 32 * 32 * sizeof(float); // 32 KB (merge only)
  if (useCache) {
    gemm_stats_kernel<true,  true><<<grid, 256, ldsS, stream>>>(s_in, s_w, s_b, HS, sM1, sS1, sM2, sS2, sLog);
    gemm_stats_kernel<false, true><<<grid, 256, ldsM, stream>>>(t_in, t_w, t_b, HT, tM1, tS1, tM2, tS2, tLog);
  } else {
    gemm_stats_kernel<true,  false><<<grid, 256, ldsS, stream>>>(s_in, s_w, s_b, HS, sM1, sS1, sM2, sS2, nullptr);
    gemm_stats_kernel<false, false><<<grid, 256, ldsM, stream>>>(t_in, t_w, t_b, HT, tM1, tS1, tM2, tS2, nullptr);
  }

  reduce_lse_kernel<<<(N_TOK + 255) / 256, 256, 0, stream>>>(sM1, sS1, lse1s);
  reduce_lse_kernel<<<(N_TOK + 255) / 256, 256, 0, stream>>>(sM2, sS2, lse2s);
  reduce_lse_kernel<<<(N_TOK + 255) / 256, 256, 0, stream>>>(tM2, tS2, lse2t);

  ce_kernel<<<N_TOK, 256, 0, stream>>>(s_in, s_w, s_b, tgt, lse1s, acc + 0);

  if (useCache) {
    jsd_cached_kernel<<<N_TOK, 256, 0, stream>>>(sLog, tLog, lse2s, lse2t, acc + 1);
  } else {
    jsd_kernel<<<grid, 256, ldsS, stream>>>(s_in, s_w, s_b, t_in, t_w, t_b, lse2s, lse2t, acc + 1);
  }

  finalize_kernel<<<1, 1, 0, stream>>>(acc, (float*)d_out);
}